// DiffusionModel_15831249453195
// MI455X (gfx1250) — compile-verified
//
#include <hip/hip_runtime.h>
#include <cmath>

// Problem constants (match reference)
#define B_  8
#define N_  2048
#define H_  256
#define THREADS 256          // 8 waves of 32
#define ROWS_PER_BLOCK 128   // 8 waves x 16 rows

typedef __attribute__((ext_vector_type(2))) float v2f;
typedef __attribute__((ext_vector_type(8))) float v8f;

#define LOG2E 1.4426950408889634f
#define LN2   0.6931471805599453f

// ---------------------------------------------------------------------------
// Prep: per-batch std (ddof=1), build padded point arrays with |p|^2 in .w,
// zero the dual potentials f,g.
// ---------------------------------------------------------------------------
__global__ __launch_bounds__(THREADS)
void prep_kernel(const float* __restrict__ cloud, const float* __restrict__ noise,
                 float4* __restrict__ ypts, float4* __restrict__ xpts,
                 float* __restrict__ f, float* __restrict__ g) {
  const int b   = blockIdx.x;
  const int tid = threadIdx.x;
  __shared__ float s_sum[THREADS];
  __shared__ float s_sq[THREADS];

  const float* cb = cloud + (size_t)b * N_ * 3;
  const float* nb = noise + (size_t)b * N_ * 3;

  float sum = 0.f, sq = 0.f;
  for (int i = tid; i < N_ * 3; i += THREADS) {
    float v = cb[i];
    sum += v; sq += v * v;
  }
  s_sum[tid] = sum; s_sq[tid] = sq;
  __syncthreads();
  for (int off = THREADS / 2; off > 0; off >>= 1) {
    if (tid < off) { s_sum[tid] += s_sum[tid + off]; s_sq[tid] += s_sq[tid + off]; }
    __syncthreads();
  }
  const float n    = (float)(N_ * 3);
  const float tot  = s_sum[0];
  const float tot2 = s_sq[0];
  const float var  = (tot2 - tot * tot / n) / (n - 1.0f);   // ddof = 1
  const float inv_std = rsqrtf(var);

  for (int p = tid; p < N_; p += THREADS) {
    float yx = cb[p * 3 + 0] * inv_std;
    float yy = cb[p * 3 + 1] * inv_std;
    float yz = cb[p * 3 + 2] * inv_std;
    ypts[(size_t)b * N_ + p] = make_float4(yx, yy, yz, yx * yx + yy * yy + yz * yz);
    float xx = nb[p * 3 + 0];
    float xy = nb[p * 3 + 1];
    float xz = nb[p * 3 + 2];
    xpts[(size_t)b * N_ + p] = make_float4(xx, xy, xz, xx * xx + xy * xy + xz * xz);
    f[(size_t)b * N_ + p] = 0.f;
    g[(size_t)b * N_ + p] = 0.f;
  }
}

// ---------------------------------------------------------------------------
// One Sinkhorn half-step:
//   out_i = -eps*(lconst + logsumexp_j((gy_j + x_i.y_j)/eps)) + 0.5*|row_i|^2
// gy_j = in_pot[j] - 0.5*|col_j|^2.  x_i.y_j tiles via WMMA f32 16x16x4.
// Online logsumexp kept in base-2 with ONE native v_exp_f32 per element:
//   d = v - m, e = exp2(-|d|):  d>0 -> (s*e+1, m=v) ; else (s+e, m).
// ---------------------------------------------------------------------------
__global__ __launch_bounds__(THREADS)
void sinkhorn_half(const float4* __restrict__ rows_pts,
                   const float4* __restrict__ cols_pts,
                   const float*  __restrict__ in_pot,
                   float*        __restrict__ out_pot,
                   float eps, float lconst) {
  const int b    = blockIdx.y;
  const int tid  = threadIdx.x;
  const int lane = tid & 31;
  const int wave = tid >> 5;
  const int hi   = lane >> 4;    // 0: K=0,1  1: K=2,3 (and rows/cols +8 in C/D)
  const int idx  = lane & 15;

  __shared__ float4 s_cols[N_];  // 32 KB
  __shared__ float  s_gy[N_];    // 8 KB

  const float4* cp = cols_pts + (size_t)b * N_;
  const float*  ip = in_pot   + (size_t)b * N_;
  for (int j = tid; j < N_; j += THREADS) {
    float4 c = cp[j];
    s_cols[j] = c;
    s_gy[j]   = ip[j] - 0.5f * c.w;
  }
  __syncthreads();

  const int row_base = blockIdx.x * ROWS_PER_BLOCK + wave * 16;
  const float4* rp = rows_pts + (size_t)b * N_;

  // A fragment: 16x4 (K padded 3->4 with 0)
  float4 r4 = rp[row_base + idx];
  v2f a;
  a.x = hi ? r4.z : r4.x;
  a.y = hi ? 0.0f : r4.y;

  float m[8], s[8];
#pragma unroll
  for (int r = 0; r < 8; ++r) { m[r] = -3.0e38f; s[r] = 0.f; }
  const float c2 = LOG2E / eps;     // fold 1/eps and ln->log2 conversion

  for (int c0 = 0; c0 < N_; c0 += 16) {
    float4 c4 = s_cols[c0 + idx];
    v2f bf;
    bf.x = hi ? c4.z : c4.x;
    bf.y = hi ? 0.0f : c4.y;
    float gy = s_gy[c0 + idx];

    v8f cz = {};
    v8f acc = __builtin_amdgcn_wmma_f32_16x16x4_f32(
        false, a, false, bf, (short)0, cz, false, false);

#pragma unroll
    for (int r = 0; r < 8; ++r) {
      float v = (gy + acc[r]) * c2;                 // base-2 log domain
      float d = v - m[r];
      float e = __builtin_amdgcn_exp2f(-fabsf(d));  // v_exp_f32 w/ neg|abs| mods
      bool  up = d > 0.0f;
      s[r] = up ? fmaf(s[r], e, 1.0f) : (s[r] + e);
      m[r] = up ? v : m[r];
    }
  }

  // combine the 16 column-lanes of each half-wave (base-2 LSE tree, 1 exp/step)
#pragma unroll
  for (int mask = 1; mask < 16; mask <<= 1) {
#pragma unroll
    for (int r = 0; r < 8; ++r) {
      float om = __shfl_xor(m[r], mask, 32);
      float os = __shfl_xor(s[r], mask, 32);
      float d  = om - m[r];
      float e  = __builtin_amdgcn_exp2f(-fabsf(d));
      bool  up = d > 0.0f;
      s[r] = up ? fmaf(s[r], e, os) : fmaf(os, e, s[r]);
      m[r] = up ? om : m[r];
    }
  }

  if (idx < 8) {
    const int row = row_base + hi * 8 + idx;
    const int r   = idx;
    float sqrow = rp[row].w;
    float lse = LN2 * (m[r] + __builtin_amdgcn_logf(s[r]));  // back to ln domain
    out_pot[(size_t)b * N_ + row] = -eps * (lconst + lse) + 0.5f * sqrow;
  }
}

// ---------------------------------------------------------------------------
// Final: hard assignment via WMMA-tiled argmax(2*x.y + g - |y|^2), then the
// tiny MLP, writing (v_pred, v) into d_out.
// ---------------------------------------------------------------------------
__global__ __launch_bounds__(THREADS)
void assign_mlp_kernel(const float4* __restrict__ xpts,
                       const float4* __restrict__ ypts,
                       const float*  __restrict__ g,
                       const float*  __restrict__ t,
                       const float*  __restrict__ W1,
                       const float*  __restrict__ Wt,
                       const float*  __restrict__ b1,
                       const float*  __restrict__ W2,
                       const float*  __restrict__ b2,
                       float*        __restrict__ out) {
  const int b    = blockIdx.y;
  const int tid  = threadIdx.x;
  const int lane = tid & 31;
  const int wave = tid >> 5;
  const int hi   = lane >> 4;
  const int idx  = lane & 15;

  __shared__ float4 s_cols[N_];          // x0 points, 32 KB
  __shared__ float  s_gy[N_];            // g - |y|^2, 8 KB
  __shared__ float  s_W1[3 * H_];
  __shared__ float  s_Wt[H_];
  __shared__ float  s_b1[H_];
  __shared__ float  s_W2[H_ * 3];
  __shared__ float  s_b2[4];
  __shared__ int    s_idx[ROWS_PER_BLOCK];

  const float4* cp = ypts + (size_t)b * N_;
  const float*  gp = g    + (size_t)b * N_;
  for (int j = tid; j < N_; j += THREADS) {
    float4 c = cp[j];
    s_cols[j] = c;
    s_gy[j]   = gp[j] - c.w;
  }
  for (int j = tid; j < 3 * H_; j += THREADS) { s_W1[j] = W1[j]; s_W2[j] = W2[j]; }
  for (int j = tid; j < H_;     j += THREADS) { s_Wt[j] = Wt[j]; s_b1[j] = b1[j]; }
  if (tid < 3) s_b2[tid] = b2[tid];
  __syncthreads();

  const int row_base = blockIdx.x * ROWS_PER_BLOCK + wave * 16;
  const float4* rp = xpts + (size_t)b * N_;

  float4 r4 = rp[row_base + idx];
  v2f a;                               // 2*x so WMMA yields 2*x.y directly
  a.x = hi ? 2.0f * r4.z : 2.0f * r4.x;
  a.y = hi ? 0.0f        : 2.0f * r4.y;

  float best[8]; int bidx[8];
#pragma unroll
  for (int r = 0; r < 8; ++r) { best[r] = -3.0e38f; bidx[r] = 0; }

  for (int c0 = 0; c0 < N_; c0 += 16) {
    float4 c4 = s_cols[c0 + idx];
    v2f bf;
    bf.x = hi ? c4.z : c4.x;
    bf.y = hi ? 0.0f : c4.y;
    float gy = s_gy[c0 + idx];

    v8f cz = {};
    v8f acc = __builtin_amdgcn_wmma_f32_16x16x4_f32(
        false, a, false, bf, (short)0, cz, false, false);

#pragma unroll
    for (int r = 0; r < 8; ++r) {
      float v = acc[r] + gy;
      if (v > best[r]) { best[r] = v; bidx[r] = c0 + idx; }   // first-max wins
    }
  }

#pragma unroll
  for (int mask = 1; mask < 16; mask <<= 1) {
#pragma unroll
    for (int r = 0; r < 8; ++r) {
      float ov = __shfl_xor(best[r], mask, 32);
      int   oi = __shfl_xor(bidx[r], mask, 32);
      if (ov > best[r] || (ov == best[r] && oi < bidx[r])) {
        best[r] = ov; bidx[r] = oi;
      }
    }
  }

  if (idx < 8) s_idx[wave * 16 + hi * 8 + idx] = bidx[idx];
  __syncthreads();

  // MLP: one thread per row (threads 0..127)
  if (tid < ROWS_PER_BLOCK) {
    const int grow = blockIdx.x * ROWS_PER_BLOCK + tid;
    const int j    = s_idx[tid];
    float4 x0a = s_cols[j];
    float4 xn  = rp[grow];
    const float tv = t[b], omt = 1.0f - tv;

    float vx = xn.x - x0a.x, vy = xn.y - x0a.y, vz = xn.z - x0a.z;
    float xtx = omt * x0a.x + tv * xn.x;
    float xty = omt * x0a.y + tv * xn.y;
    float xtz = omt * x0a.z + tv * xn.z;

    float p0 = s_b2[0], p1 = s_b2[1], p2 = s_b2[2];
    for (int k = 0; k < H_; ++k) {
      float h = xtx * s_W1[k] + xty * s_W1[H_ + k] + xtz * s_W1[2 * H_ + k]
              + tv * s_Wt[k] + s_b1[k];
      h = fmaxf(h, 0.0f);
      p0 += h * s_W2[k * 3 + 0];
      p1 += h * s_W2[k * 3 + 1];
      p2 += h * s_W2[k * 3 + 2];
    }
    const size_t base  = ((size_t)b * N_ + grow) * 3;
    const size_t vbase = (size_t)B_ * N_ * 3 + base;
    out[base + 0] = p0; out[base + 1] = p1; out[base + 2] = p2;
    out[vbase + 0] = vx; out[vbase + 1] = vy; out[vbase + 2] = vz;
  }
}

// ---------------------------------------------------------------------------
extern "C" void kernel_launch(void* const* d_in, const int* in_sizes, int n_in,
                              void* d_out, int out_size, void* d_ws, size_t ws_size,
                              hipStream_t stream) {
  (void)in_sizes; (void)n_in; (void)out_size; (void)ws_size;
  const float* cloud = (const float*)d_in[0];
  const float* noise = (const float*)d_in[1];
  const float* t     = (const float*)d_in[2];
  const float* W1    = (const float*)d_in[3];
  const float* Wt    = (const float*)d_in[4];
  const float* b1    = (const float*)d_in[5];
  const float* W2    = (const float*)d_in[6];
  const float* b2    = (const float*)d_in[7];
  float* out = (float*)d_out;

  char* ws = (char*)d_ws;
  float4* ypts = (float4*)ws;                                   // B*N float4
  float4* xpts = (float4*)(ws + (size_t)B_ * N_ * sizeof(float4));
  float*  f    = (float*) (ws + (size_t)2 * B_ * N_ * sizeof(float4));
  float*  g    = f + (size_t)B_ * N_;

  prep_kernel<<<B_, THREADS, 0, stream>>>(cloud, noise, ypts, xpts, f, g);

  const float lconst = -logf((float)N_);                        // loga == logb
  dim3 grid(N_ / ROWS_PER_BLOCK, B_);
  for (int it = 0; it < 14; ++it) {
    // geomspace(32.0, 1e-6, 14)
    float eps = (float)(32.0 * pow(1e-6 / 32.0, (double)it / 13.0));
    sinkhorn_half<<<grid, THREADS, 0, stream>>>(xpts, ypts, g, f, eps, lconst);
    sinkhorn_half<<<grid, THREADS, 0, stream>>>(ypts, xpts, f, g, eps, lconst);
  }

  assign_mlp_kernel<<<grid, THREADS, 0, stream>>>(xpts, ypts, g, t,
                                                  W1, Wt, b1, W2, b2, out);
}